// PositionalEmbedding_82042465288206
// MI455X (gfx1250) — compile-verified
//
#include <hip/hip_runtime.h>

// PositionalEmbedding generator for MI455X (gfx1250).
// out[b, s, d] = sin/cos((start_pos + s) * 10000^(-2d/1024)), b broadcast.
//
// Strategy:
//  - angle = pos (.) inv_freq outer product via V_WMMA_F32_16X16X4_F32 (K=0 only).
//  - one v_sin_f32 per element using cos(x) = sin(x + 2*pi*0.25) on odd lanes.
//  - 16x1024 f32 tile (64 KiB) staged in LDS, replicated to 4 batch copies with
//    GLOBAL_STORE_ASYNC_FROM_LDS_B128 (ASYNCcnt) -> store-bandwidth-bound (~5.6us floor).

typedef __attribute__((ext_vector_type(2))) float v2f;
typedef __attribute__((ext_vector_type(8))) float v8f;
typedef __attribute__((ext_vector_type(4))) int   v4i;

#define S_TOTAL 8192
#define D_TOTAL 1024
#define B_TOTAL 4
#define ROWS    16          // positions per block == WMMA M

__global__ __launch_bounds__(256) void pe_wmma_kernel(const int* __restrict__ start_pos_p,
                                                      float* __restrict__ out) {
    __shared__ float tile[ROWS * D_TOTAL];   // 64 KiB, contiguous image of 16 output rows

    const int tid     = threadIdx.x;         // 0..255
    const int lane    = tid & 31;            // wave32
    const int wave    = tid >> 5;            // 0..7
    const int rowBase = blockIdx.x * ROWS;
    const int sp      = start_pos_p[0];

    const int   half    = (lane >= 16) ? 8 : 0;                 // C/D layout: lanes 16..31 -> M += 8
    const float sel     = (lane & 1) ? 0.25f : 0.0f;            // cos via quarter-revolution shift
    const float inv2pi  = 0.15915494309189535f;                 // v_sin takes revolutions
    const float negc    = -0.025952563241282f;                  // -log2(10000)/512

    // A (16x4 f32): column K=0 = positions, K=1..3 = 0. Lanes 0-15 hold K=0,1; lanes 16-31 K=2,3.
    v2f a;
    a.x = (lane < 16) ? (float)(sp + rowBase + lane) : 0.0f;
    a.y = 0.0f;

    // Each wave produces 8 16x16 tiles covering cols [wave*128, wave*128+128)
    for (int t = 0; t < 8; ++t) {
        const int colBase = (wave * 8 + t) * 16;
        const int n       = lane & 15;

        // B (4x16 f32): row K=0 = inv_freq(colBase + n) on lanes 0-15.
        v2f b;
        b.x = (lane < 16) ? __builtin_amdgcn_exp2f((float)(colBase + n) * negc) : 0.0f;
        b.y = 0.0f;

        v8f acc;
#if __has_builtin(__builtin_amdgcn_wmma_f32_16x16x4_f32)
        v8f c = {};
        acc = __builtin_amdgcn_wmma_f32_16x16x4_f32(
                  /*neg_a=*/false, a, /*neg_b=*/false, b,
                  /*c_mod=*/(short)0, c, /*reuse_a=*/false, /*reuse_b=*/false);
#else
        // Fallback: scalar outer product (should not be needed on gfx1250)
        #pragma unroll
        for (int v = 0; v < 8; ++v)
            acc[v] = (float)(sp + rowBase + v + half) *
                     __builtin_amdgcn_exp2f((float)(colBase + n) * negc);
#endif

        #pragma unroll
        for (int v = 0; v < 8; ++v) {
            // revolutions + parity shift, range-reduce, single hw sine
            float r = acc[v] * inv2pi + sel;
            r -= floorf(r);
            float val = __builtin_amdgcn_sinf(r);
            const int m = v + half;
            tile[m * D_TOTAL + colBase + n] = val;
        }
    }

    __syncthreads();

    // Replicate the 64 KiB tile to all 4 batch images via async LDS->global DMA stores.
#if __has_builtin(__builtin_amdgcn_global_store_async_from_lds_b128)
    typedef __attribute__((address_space(1))) char  gchar;
    typedef __attribute__((address_space(3))) char  lchar;
    typedef __attribute__((address_space(1))) v4i   gv4i;
    typedef __attribute__((address_space(3))) v4i   lv4i;
    lchar* ldsb = (lchar*)tile;
    for (int bb = 0; bb < B_TOTAL; ++bb) {
        gchar* dstb = (gchar*)(out + ((size_t)bb * S_TOTAL + rowBase) * D_TOTAL);
        for (int i = 0; i < 16; ++i) {
            const int off = (i * 256 + tid) * 16;   // 16 B per lane per op, 64 KiB total
            __builtin_amdgcn_global_store_async_from_lds_b128(
                (gv4i*)(dstb + off), (lv4i*)(ldsb + off), 0, 0);
        }
    }
#if __has_builtin(__builtin_amdgcn_s_wait_asynccnt)
    __builtin_amdgcn_s_wait_asynccnt(0);
#endif
#else
    // Fallback: 128-bit register stores from LDS
    const float4* lv = (const float4*)tile;
    for (int i = 0; i < 16; ++i) {
        const int idx = i * 256 + tid;              // float4 index, 4096 total
        float4 vdat = lv[idx];
        #pragma unroll
        for (int bb = 0; bb < B_TOTAL; ++bb)
            ((float4*)(out + ((size_t)bb * S_TOTAL + rowBase) * D_TOTAL))[idx] = vdat;
    }
#endif
}

extern "C" void kernel_launch(void* const* d_in, const int* in_sizes, int n_in,
                              void* d_out, int out_size, void* d_ws, size_t ws_size,
                              hipStream_t stream) {
    (void)in_sizes; (void)n_in; (void)d_ws; (void)ws_size; (void)out_size;
    // d_in[0] = x (shape donor only, unused), d_in[1] = start_pos (scalar int)
    const int* start_pos = (const int*)d_in[1];
    float* out = (float*)d_out;

    dim3 grid(S_TOTAL / ROWS);   // 512 blocks
    dim3 block(256);             // 8 wave32s
    pe_wmma_kernel<<<grid, block, 0, stream>>>(start_pos, out);
}